// window_embeddings_65377992179823
// MI455X (gfx1250) — compile-verified
//
#include <hip/hip_runtime.h>
#include <math.h>

#define B_ 8
#define W_ 64
#define N_ 128
#define D_ 1024
#define P_ 512
#define V_ 256
#define NWIN (B_ * W_)         // 512 windows
#define OUTD (D_ + V_)         // 1280

typedef __attribute__((ext_vector_type(2))) float v2f;
typedef __attribute__((ext_vector_type(4))) float v4f;   // native vector: OK for NT builtins
typedef __attribute__((ext_vector_type(8))) float v8f;

// ---------------------------------------------------------------------------
// Kernel 1: per-window masked pooling + copy x into out[..., :D]
// One block (256 threads) per window; v4f (B128) streaming, NT stores for
// the output copy so the 320MB 'out' stream does not evict x from the 192MB L2.
// ---------------------------------------------------------------------------
__global__ __launch_bounds__(256) void k_pool_copy(
    const float* __restrict__ x, const unsigned char* __restrict__ mask,
    float* __restrict__ out, float* __restrict__ pooled,
    float* __restrict__ inv_len) {
  const int win = blockIdx.x;
  const int t = threadIdx.x;
  __shared__ int s_mask[N_];
  __shared__ int s_cnt;
  if (t == 0) s_cnt = 0;
  __syncthreads();
  if (t < N_) {
    int m = mask[win * N_ + t] ? 1 : 0;
    s_mask[t] = m;
    if (!m) atomicAdd(&s_cnt, 1);
  }
  __syncthreads();

  const v4f* xw4 = (const v4f*)(x + (size_t)win * N_ * D_);
  float* ow = out + (size_t)win * N_ * OUTD;
  v4f acc = {0.f, 0.f, 0.f, 0.f};
  for (int n = 0; n < N_; ++n) {
    v4f v = xw4[(size_t)n * (D_ / 4) + t];              // coalesced 512B/wave
    v4f* orow4 = (v4f*)(ow + (size_t)n * OUTD);
    __builtin_nontemporal_store(v, orow4 + t);          // NT: out never re-read
    if (!s_mask[n]) acc += v;
  }
  ((v4f*)(pooled + (size_t)win * D_))[t] = acc;
  if (t == 0) {
    int c = s_cnt; if (c < 1) c = 1;
    inv_len[win] = 1.0f / (float)c;
  }
}

// ---------------------------------------------------------------------------
// Kernel: sinusoidal positional table POS[N_][P_], scaled by pos_scale
// ---------------------------------------------------------------------------
__global__ void k_pos(float* __restrict__ pos, const float* __restrict__ pos_scale) {
  int i = blockIdx.x * blockDim.x + threadIdx.x;
  if (i >= N_ * P_) return;
  int j = i / P_;
  int p = i % P_;
  int m = (p < P_ / 2) ? p : (p - P_ / 2);
  // inv_freq = 10000 ^ (-(2m)/P)
  float invf = __expf(-((float)(2 * m) / (float)P_) * 9.210340371976184f);
  float ang = (float)j * invf;
  float val = (p < P_ / 2) ? sinf(ang) : cosf(ang);
  pos[i] = val * pos_scale[0];
}

// ---------------------------------------------------------------------------
// Generic f32 WMMA GEMM: C[M,N] = A[M,K] @ B  (B row-major [K,N], or if
// btrans: B is [N,K] row-major and we use B^T). Epilogue: optional per-row
// scale and ReLU. One 16x16 C tile per wave, V_WMMA_F32_16X16X4_F32 K-loop.
// ---------------------------------------------------------------------------
__global__ __launch_bounds__(256) void k_gemm_wmma(
    const float* __restrict__ A, int lda,
    const float* __restrict__ B, int ldb, int btrans,
    float* __restrict__ C, int ldc,
    int M, int N, int K,
    const float* __restrict__ rowscale, int do_relu) {
  const int wave = threadIdx.x >> 5;
  const int lane = threadIdx.x & 31;
  const int tiles_n = N >> 4;
  const int tiles_m = M >> 4;
  const int tile = blockIdx.x * 8 + wave;
  if (tile >= tiles_m * tiles_n) return;   // whole-wave uniform exit
  const int tm = tile / tiles_n;
  const int tn = tile - tm * tiles_n;
  const int half = lane >> 4;   // 0: K pair {0,1}, 1: K pair {2,3}
  const int idx = lane & 15;    // A row within tile / B+C col within tile

  const float* Arow = A + (size_t)(tm * 16 + idx) * lda;
  v8f c = {};
  if (!btrans) {
    const float* Bcol = B + (tn * 16 + idx);
#pragma unroll 4
    for (int k0 = 0; k0 < K; k0 += 4) {
      int kk = k0 + 2 * half;
      v2f a, b;
      a[0] = Arow[kk];
      a[1] = Arow[kk + 1];
      b[0] = Bcol[(size_t)kk * ldb];
      b[1] = Bcol[(size_t)(kk + 1) * ldb];
      c = __builtin_amdgcn_wmma_f32_16x16x4_f32(false, a, false, b,
                                                (short)0, c, false, false);
    }
  } else {
    const float* Brow = B + (size_t)(tn * 16 + idx) * ldb;
#pragma unroll 4
    for (int k0 = 0; k0 < K; k0 += 4) {
      int kk = k0 + 2 * half;
      v2f a, b;
      a[0] = Arow[kk];
      a[1] = Arow[kk + 1];
      b[0] = Brow[kk];
      b[1] = Brow[kk + 1];
      c = __builtin_amdgcn_wmma_f32_16x16x4_f32(false, a, false, b,
                                                (short)0, c, false, false);
    }
  }
#pragma unroll
  for (int r = 0; r < 8; ++r) {
    int row = tm * 16 + r + 8 * half;   // C layout: VGPR r -> M=r (+8 for hi half)
    int col = tn * 16 + idx;
    float v = c[r];
    if (rowscale) v *= rowscale[row];
    if (do_relu) v = fmaxf(v, 0.0f);
    C[(size_t)row * ldc + col] = v;
  }
}

// ---------------------------------------------------------------------------
// Kernel: per-window attention. dots_j = (x_j . qk_win + qpos) * P^-0.5,
// mask, softmax, then pooled2 = attn @ x_window. One block per window.
// x window (512KB) is re-read from L2 between phases.
// ---------------------------------------------------------------------------
__global__ __launch_bounds__(256) void k_attend(
    const float* __restrict__ x, const unsigned char* __restrict__ mask,
    const float* __restrict__ qk, const float* __restrict__ qpos,
    float* __restrict__ pooled2) {
  const int win = blockIdx.x;
  const int t = threadIdx.x;
  const int wave = t >> 5;
  const int lane = t & 31;
  __shared__ float s_dots[N_];
  __shared__ float s_red[256];

  const v4f* xw4 = (const v4f*)(x + (size_t)win * N_ * D_);
  const v4f* qk4 = (const v4f*)(qk + (size_t)win * D_);

  // Phase 1: dots via coalesced v4f GEMV, wave-shuffle reduction
  for (int jj = 0; jj < N_ / 8; ++jj) {
    int j = wave + jj * 8;
    const v4f* xr4 = xw4 + (size_t)j * (D_ / 4);
    float acc = 0.f;
#pragma unroll
    for (int d = lane; d < D_ / 4; d += 32) {
      v4f a = xr4[d];
      v4f b = qk4[d];
      acc += a.x * b.x + a.y * b.y + a.z * b.z + a.w * b.w;
    }
    for (int off = 16; off > 0; off >>= 1) acc += __shfl_xor(acc, off, 32);
    if (lane == 0) s_dots[j] = acc;
  }
  __syncthreads();

  // Phase 2: add pos term, scale, mask
  if (t < N_) {
    float dd = (s_dots[t] + qpos[(size_t)win * N_ + t]) * 0.04419417382415922f; // 512^-0.5
    if (mask[win * N_ + t]) dd = -3.402823466e38f;
    s_dots[t] = dd;
  }
  __syncthreads();

  // max-reduce
  s_red[t] = (t < N_) ? s_dots[t] : -3.402823466e38f;
  __syncthreads();
  for (int s = 128; s > 0; s >>= 1) {
    if (t < s) s_red[t] = fmaxf(s_red[t], s_red[t + s]);
    __syncthreads();
  }
  float mx = s_red[0];
  __syncthreads();

  // exp + sum-reduce
  float e = 0.f;
  if (t < N_) { e = __expf(s_dots[t] - mx); s_dots[t] = e; }
  s_red[t] = (t < N_) ? e : 0.f;
  __syncthreads();
  for (int s = 128; s > 0; s >>= 1) {
    if (t < s) s_red[t] += s_red[t + s];
    __syncthreads();
  }
  float inv_sum = 1.0f / s_red[0];
  __syncthreads();
  if (t < N_) s_dots[t] *= inv_sum;
  __syncthreads();

  // Phase 3: pooled2[d] = sum_j attn_j * x[j,d]  (v4f, L2-resident)
  v4f acc = {0.f, 0.f, 0.f, 0.f};
  for (int j = 0; j < N_; ++j) {
    float wgt = s_dots[j];
    v4f v = xw4[(size_t)j * (D_ / 4) + t];
    acc += wgt * v;
  }
  ((v4f*)(pooled2 + (size_t)win * D_))[t] = acc;
}

// ---------------------------------------------------------------------------
// Kernel: scatter attention outputs with the bos shift into out[..., D:]
// out_used[b, w] = bos if w==0 else outv[b, w-1], broadcast over n.
// v4f NT stores (write-once stream).
// ---------------------------------------------------------------------------
__global__ __launch_bounds__(256) void k_writeout(
    const float* __restrict__ outv, const float* __restrict__ bos,
    float* __restrict__ out) {
  int gid = blockIdx.x * 256 + threadIdx.x;   // over NWIN * N_ * (V_/4)
  int v4 = gid & (V_ / 4 - 1);                // 0..63
  int n = (gid >> 6) & (N_ - 1);
  int win = gid >> 13;                        // 0..511
  int w = win & (W_ - 1);
  const v4f* src4 = (w == 0) ? (const v4f*)bos
                             : (const v4f*)(outv + (size_t)(win - 1) * V_);
  v4f val = src4[v4];
  v4f* dst = (v4f*)(out + (size_t)win * N_ * OUTD + (size_t)n * OUTD + D_);
  __builtin_nontemporal_store(val, dst + v4);
}

// ---------------------------------------------------------------------------
extern "C" void kernel_launch(void* const* d_in, const int* in_sizes, int n_in,
                              void* d_out, int out_size, void* d_ws, size_t ws_size,
                              hipStream_t stream) {
  const float* x = (const float*)d_in[0];
  const unsigned char* mask = (const unsigned char*)d_in[1];  // numpy bool layout
  const float* W_in = (const float*)d_in[2];   // [D, P]
  const float* W_q  = (const float*)d_in[3];   // [P, P]
  const float* W_k  = (const float*)d_in[4];   // [D, P]
  const float* W_v  = (const float*)d_in[5];   // [D, V]
  const float* pos_scale = (const float*)d_in[6];
  const float* bos  = (const float*)d_in[7];   // [V]
  float* out = (float*)d_out;

  // Workspace layout (floats)
  float* ws = (float*)d_ws;
  float* POS     = ws;                 size_t o = (size_t)N_ * P_;      // 65536
  float* pooled  = ws + o;             o += (size_t)NWIN * D_;          // 524288
  float* invlen  = ws + o;             o += NWIN;                       // 512
  float* WE      = ws + o;             o += (size_t)NWIN * P_;          // 262144
  float* Q       = ws + o;             o += (size_t)NWIN * P_;          // 262144
  float* QK      = ws + o;             o += (size_t)NWIN * D_;          // 524288
  float* QPOS    = ws + o;             o += (size_t)NWIN * N_;          // 65536
  float* pooled2 = ws + o;             o += (size_t)NWIN * D_;          // 524288
  float* OUTV    = ws + o;             o += (size_t)NWIN * V_;          // 131072

  // 1) pool + copy x into out[..., :D]
  k_pool_copy<<<NWIN, 256, 0, stream>>>(x, mask, out, pooled, invlen);

  // 2) positional table
  k_pos<<<(N_ * P_ + 255) / 256, 256, 0, stream>>>(POS, pos_scale);

  // 3) WMMA GEMM chain (batched over all 512 windows)
  // WE = relu((pooled @ W_in) * invlen)   [512,512]
  {
    int M = NWIN, N = P_, K = D_;
    int tiles = (M / 16) * (N / 16);
    k_gemm_wmma<<<(tiles + 7) / 8, 256, 0, stream>>>(
        pooled, D_, W_in, P_, 0, WE, P_, M, N, K, invlen, 1);
  }
  // Q = WE @ W_q                          [512,512]
  {
    int M = NWIN, N = P_, K = P_;
    int tiles = (M / 16) * (N / 16);
    k_gemm_wmma<<<(tiles + 7) / 8, 256, 0, stream>>>(
        WE, P_, W_q, P_, 0, Q, P_, M, N, K, nullptr, 0);
  }
  // QK = Q @ W_k^T                        [512,1024]
  {
    int M = NWIN, N = D_, K = P_;
    int tiles = (M / 16) * (N / 16);
    k_gemm_wmma<<<(tiles + 7) / 8, 256, 0, stream>>>(
        Q, P_, W_k, P_, 1, QK, D_, M, N, K, nullptr, 0);
  }
  // QPOS = Q @ POS^T                      [512,128]
  {
    int M = NWIN, N = N_, K = P_;
    int tiles = (M / 16) * (N / 16);
    k_gemm_wmma<<<(tiles + 7) / 8, 256, 0, stream>>>(
        Q, P_, POS, P_, 1, QPOS, N_, M, N, K, nullptr, 0);
  }

  // 4) per-window attention -> pooled2
  k_attend<<<NWIN, 256, 0, stream>>>(x, mask, QK, QPOS, pooled2);

  // 5) OUTV = pooled2 @ W_v               [512,256]
  {
    int M = NWIN, N = V_, K = D_;
    int tiles = (M / 16) * (N / 16);
    k_gemm_wmma<<<(tiles + 7) / 8, 256, 0, stream>>>(
        pooled2, D_, W_v, V_, 0, OUTV, V_, M, N, K, nullptr, 0);
  }

  // 6) bos-shifted broadcast into out[..., D:]
  k_writeout<<<(NWIN * N_ * V_ / 4) / 256, 256, 0, stream>>>(OUTV, bos, out);
}